// Top1Router_6236292514569
// MI455X (gfx1250) — compile-verified
//
#include <hip/hip_runtime.h>
#include <hip/hip_bf16.h>

// ---------------------------------------------------------------------------
// Top-1 MoE router for MI455X (gfx1250, wave32).
//   logits = hidden[16384,2048] @ W[8,2048]^T + b      (HBM-bound, ~5.8us floor)
//   max_probs = 1/sum(exp(l - max))  ; expert = argmax
//   capacity cumsum (<=640 per expert per batch) -> masked one-hot
// GEMV uses V_WMMA_F32_16X16X4_F32 (only f32 WMMA shape on CDNA5).
// ---------------------------------------------------------------------------

typedef __attribute__((ext_vector_type(2))) float v2f;
typedef __attribute__((ext_vector_type(8))) float v8f;

#define NUM_EXPERTS 8
#define HIDDEN      2048
#define BATCH       4
#define SEQ         4096
#define TOKENS      (BATCH * SEQ)
#define CAPACITY    640
#define WAVES_PER_BLOCK 8

// ---------------------------------------------------------------------------
// Kernel 1: one wave computes a 16-token x 16-col (8 valid experts) logit tile
// via 512 chained V_WMMA_F32_16X16X4_F32, then per-token softmax-max + argmax.
// ---------------------------------------------------------------------------
__global__ __launch_bounds__(256) void router_gemv_wmma(
    const float* __restrict__ hidden,   // [TOKENS, HIDDEN]
    const float* __restrict__ W,        // [NUM_EXPERTS, HIDDEN]
    const float* __restrict__ bias,     // [NUM_EXPERTS]
    float* __restrict__ out_logits,     // [TOKENS, NUM_EXPERTS]
    float* __restrict__ out_maxprob,    // [TOKENS]
    int*   __restrict__ expert_id)      // [TOKENS] workspace
{
    __shared__ float lds_logits[WAVES_PER_BLOCK][16][NUM_EXPERTS];

    const int lane  = threadIdx.x & 31;
    const int wave  = threadIdx.x >> 5;
    const int tile  = blockIdx.x * WAVES_PER_BLOCK + wave;
    const int token0 = tile * 16;

    // WMMA f32 16x16x4 fragment layout (ISA 7.12.2):
    //  A (16x4): lanes 0-15 -> row=lane, VGPR{0,1}=K{0,1}; lanes 16-31 -> K{2,3}
    //  B (4x16): transpose-symmetric: col=lane&15, same K split
    //  C (16x16): VGPR j -> M=j (lanes 0-15) / M=j+8 (lanes 16-31), N=lane&15
    const int half  = lane >> 4;          // 0 -> K{0,1}, 1 -> K{2,3}
    const int rc    = lane & 15;          // A row within tile / B column
    const int kbase = half * 2;

    const float* Arow = hidden + (size_t)(token0 + rc) * HIDDEN + kbase;
    // Columns 8..15 alias experts 0..7: their D columns are never read and the
    // values stay finite (no NaN injection), and the duplicate loads coalesce.
    const float* Brow = W + (size_t)(rc & (NUM_EXPERTS - 1)) * HIDDEN + kbase;

    v8f c = {0.f, 0.f, 0.f, 0.f, 0.f, 0.f, 0.f, 0.f};

    #pragma unroll 4
    for (int k = 0; k < HIDDEN; k += 4) {
        v2f a = *(const v2f*)(Arow + k);   // 8B contiguous per lane
        v2f b = *(const v2f*)(Brow + k);
        c = __builtin_amdgcn_wmma_f32_16x16x4_f32(
                /*neg_a=*/false, a, /*neg_b=*/false, b,
                /*c_mod=*/(short)0, c, /*reuse_a=*/false, /*reuse_b=*/false);
    }

    // Add router bias (per expert column).
    const float bn = bias[rc & (NUM_EXPERTS - 1)];
    #pragma unroll
    for (int j = 0; j < 8; ++j) c[j] += bn;

    // Scatter valid columns (experts 0..7) to LDS + global router_logits.
    const int mrow0 = half * 8;
    if (rc < NUM_EXPERTS) {
        #pragma unroll
        for (int j = 0; j < 8; ++j) {
            const int m = mrow0 + j;
            lds_logits[wave][m][rc] = c[j];
            out_logits[(size_t)(token0 + m) * NUM_EXPERTS + rc] = c[j];
        }
    }
    __syncthreads();

    // Per-token reduction over the 8 experts: lanes 0..15 each own one token.
    if (lane < 16) {
        const int token = token0 + lane;
        float l0 = lds_logits[wave][lane][0];
        float mx = l0;
        int arg = 0;
        #pragma unroll
        for (int e = 1; e < NUM_EXPERTS; ++e) {
            const float l = lds_logits[wave][lane][e];
            if (l > mx) { mx = l; arg = e; }   // first-max, matches jnp.argmax
        }
        float s = 0.f;
        #pragma unroll
        for (int e = 0; e < NUM_EXPERTS; ++e)
            s += __expf(lds_logits[wave][lane][e] - mx);
        out_maxprob[token] = 1.0f / s;         // max softmax prob = exp(0)/sum
        expert_id[token] = arg;
    }
}

// ---------------------------------------------------------------------------
// Kernel 2: per-batch capacity scan. One block per batch, 8 waves. Ballot-based
// per-expert prefix within each 256-token chunk, LDS scan across waves, running
// per-expert base across the 16 chunks.
// ---------------------------------------------------------------------------
__global__ __launch_bounds__(256) void capacity_scan(
    const int* __restrict__ expert_id,        // [TOKENS]
    float* __restrict__ out_expert_index)     // [TOKENS, NUM_EXPERTS] (0/1)
{
    __shared__ int wave_tot[8][NUM_EXPERTS];
    __shared__ int wave_base[8][NUM_EXPERTS];
    __shared__ int base[NUM_EXPERTS];

    const int tid  = threadIdx.x;
    const int lane = tid & 31;
    const int wave = tid >> 5;
    const int b    = blockIdx.x;

    if (tid < NUM_EXPERTS) base[tid] = 0;
    __syncthreads();

    for (int chunk = 0; chunk < SEQ / 256; ++chunk) {
        const int tok = b * SEQ + chunk * 256 + tid;
        const int e   = expert_id[tok];

        unsigned my_mask = 0u;
        int my_tot = 0;
        #pragma unroll
        for (int q = 0; q < NUM_EXPERTS; ++q) {
            const unsigned m = __builtin_amdgcn_ballot_w32(e == q);  // wave32
            if (q == e)    my_mask = m;
            if (lane == q) my_tot  = __popc(m);
        }
        if (lane < NUM_EXPERTS) wave_tot[wave][lane] = my_tot;
        __syncthreads();

        if (tid < 64) {  // exclusive scan of wave totals (8 waves x 8 experts)
            const int w = tid >> 3, q = tid & 7;
            int acc = 0;
            for (int ww = 0; ww < w; ++ww) acc += wave_tot[ww][q];
            wave_base[w][q] = acc;
        }
        __syncthreads();

        const int prefix = __popc(my_mask & ((1u << lane) - 1u));
        const int prio   = base[e] + wave_base[wave][e] + prefix + 1; // inclusive
        const int keep   = (prio <= CAPACITY) ? 1 : 0;

        float* o = out_expert_index + (size_t)tok * NUM_EXPERTS;
        #pragma unroll
        for (int j = 0; j < NUM_EXPERTS; ++j)
            o[j] = (j == e && keep) ? 1.0f : 0.0f;
        __syncthreads();

        if (tid < NUM_EXPERTS) {   // advance running per-expert counts
            int acc = 0;
            for (int w = 0; w < 8; ++w) acc += wave_tot[w][tid];
            base[tid] += acc;
        }
        __syncthreads();
    }
}

// ---------------------------------------------------------------------------
extern "C" void kernel_launch(void* const* d_in, const int* in_sizes, int n_in,
                              void* d_out, int out_size, void* d_ws, size_t ws_size,
                              hipStream_t stream) {
    const float* hidden = (const float*)d_in[0];   // [4,4096,2048] f32
    const float* W      = (const float*)d_in[1];   // [8,2048] f32
    const float* bias   = (const float*)d_in[2];   // [8] f32

    // d_out = expert_index(131072) ++ max_probs(16384) ++ router_logits(131072)
    float* out              = (float*)d_out;
    float* out_expert_index = out;
    float* out_maxprob      = out + (size_t)TOKENS * NUM_EXPERTS;
    float* out_logits       = out_maxprob + TOKENS;

    int* expert_id = (int*)d_ws;   // 16384 ints of scratch

    const int tiles  = TOKENS / 16;                 // 1024 waves
    const int blocks = tiles / WAVES_PER_BLOCK;     // 128 blocks x 256 threads
    router_gemv_wmma<<<dim3(blocks), dim3(256), 0, stream>>>(
        hidden, W, bias, out_logits, out_maxprob, expert_id);

    capacity_scan<<<dim3(BATCH), dim3(256), 0, stream>>>(
        expert_id, out_expert_index);
}